// ProxyFusion_21809843929951
// MI455X (gfx1250) — compile-verified
//
#include <hip/hip_runtime.h>

typedef __attribute__((ext_vector_type(2))) float v2f;
typedef __attribute__((ext_vector_type(8))) float v8f;

#define DIM   512
#define KALL  11
#define KTOP  4
#define BP    128
#define LP    256
#define BG    64
#define LG    512
#define NITEMS (BP + BG)                 // 192
#define NPAIRS (NITEMS * KTOP)           // 768
#define COLPAD 832                       // padded column count for grouped GEMM buffers
#define NEG_INF_F (-1e30f)

// ---- d_out offsets (flat concat of the 4 reference outputs) ----
#define OUT_FP 0
#define OUT_PP (BP * KTOP * DIM)                    // 262144
#define OUT_FG (OUT_PP + KALL * 10)                 // 262254
#define OUT_PG (OUT_FG + BG * KTOP * DIM)           // 393326

// ---- workspace layout: ints in [0,4096), floats from index 4096 ----
#define IW_COUNTS 0
#define IW_OFFS   16
#define IW_EXPERT 32
#define IW_SLOT   (32 + NPAIRS)
#define IW_COLIDX (32 + 2 * NPAIRS)

#define F_TPP    ((size_t)4096)
#define F_TPG    (F_TPP + 128)
#define F_MEAN   (F_TPG + 128)
#define F_VAR    (F_MEAN + (size_t)NITEMS * DIM)
#define F_INVN   (F_VAR + (size_t)NITEMS * DIM)
#define F_SMAT   (F_INVN + (size_t)(BP * LP + BG * LG))
#define F_H1     (F_SMAT + (size_t)COLPAD * 1536)
#define F_H2     (F_H1 + (size_t)COLPAD * 1024)
#define F_CTX    (F_H2 + (size_t)COLPAD * 1024)
#define F_CTXP   (F_CTX + (size_t)COLPAD * 512)
#define F_CTILDE (F_CTXP + (size_t)BP * KTOP * DIM)
#define F_BETA   (F_CTILDE + (size_t)BP * KTOP * DIM)

__device__ __forceinline__ float wred_sum(float v) {
#pragma unroll
  for (int off = 16; off >= 1; off >>= 1) v += __shfl_xor(v, off, 32);
  return v;
}
__device__ __forceinline__ float wred_max(float v) {
#pragma unroll
  for (int off = 16; off >= 1; off >>= 1) v = fmaxf(v, __shfl_xor(v, off, 32));
  return v;
}

// ---------------------------------------------------------------------------
// Kernel 0: zero counts; proxy transforms (raw -> d_out, normalized -> ws)
// ---------------------------------------------------------------------------
__global__ void k_init(const float* __restrict__ pp, const float* __restrict__ pg,
                       const float* __restrict__ tW, const float* __restrict__ tb,
                       float* __restrict__ out, float* __restrict__ wf, int* __restrict__ wi) {
  __shared__ float u[2][KALL][10];
  int tid = threadIdx.x;
  if (tid < 16) wi[IW_COUNTS + tid] = 0;
  if (tid < 220) {
    int grp = tid / 110, r = tid % 110, c = r / 10, j = r % 10;
    const float* px = grp ? pg : pp;
    const float* prow = px + (size_t)c * DIM;
    const float* wrow = tW + (size_t)j * DIM;
    float s = tb[j];
    for (int d = 0; d < DIM; ++d) s += prow[d] * wrow[d];
    u[grp][c][j] = s;
    out[(grp ? OUT_PG : OUT_PP) + c * 10 + j] = s;
  }
  __syncthreads();
  if (tid < 2 * KALL) {
    int grp = tid / KALL, c = tid % KALL;
    float n2 = 0.f;
    for (int j = 0; j < 10; ++j) n2 += u[grp][c][j] * u[grp][c][j];
    float rn = 1.0f / fmaxf(sqrtf(n2), 1e-12f);
    float* dst = wf + (grp ? F_TPG : F_TPP) + (size_t)c * 10;
    for (int j = 0; j < 10; ++j) dst[j] = u[grp][c][j] * rn;
  }
}

// ---------------------------------------------------------------------------
// Kernel 1: per-item stats: inv row norms, gsim -> top4 experts, mean/var
// ---------------------------------------------------------------------------
__global__ void __launch_bounds__(256) k_stats(
    const float* __restrict__ probes, const float* __restrict__ gallery,
    const int* __restrict__ plen, const int* __restrict__ glen,
    const float* __restrict__ tW, const float* __restrict__ tb_g,
    float* __restrict__ wf, int* __restrict__ wi) {
  __shared__ float Wl[10][DIM];
  __shared__ float tpl[KALL][10];
  __shared__ float tbl[10];
  __shared__ float gspw[8][KALL];
  __shared__ float red[8][DIM];
  __shared__ float red2[8][DIM];

  int item = blockIdx.x, tid = threadIdx.x;
  bool isP = item < BP;
  const float* feat = isP ? probes + (size_t)item * LP * DIM
                          : gallery + (size_t)(item - BP) * LG * DIM;
  int L = isP ? LP : LG;
  int length = isP ? plen[item] : glen[item - BP];
  const float* tp = wf + (isP ? F_TPP : F_TPG);
  size_t rowbase = isP ? (size_t)item * LP : (size_t)BP * LP + (size_t)(item - BP) * LG;

  for (int i = tid; i < 10 * DIM; i += 256) Wl[i / DIM][i % DIM] = tW[i];
  for (int i = tid; i < KALL * 10; i += 256) tpl[i / 10][i % 10] = tp[i];
  if (tid < 10) tbl[tid] = tb_g[tid];
  __syncthreads();

  int wave = tid >> 5, lane = tid & 31;
  int dbase = lane * 16;
  float sum16[16], sq16[16];
#pragma unroll
  for (int i = 0; i < 16; ++i) { sum16[i] = 0.f; sq16[i] = 0.f; }
  float gacc = 0.f;

  for (int l = wave; l < L; l += 8) {
    const float4* rp = (const float4*)(feat + (size_t)l * DIM + dbase);
    float4 f0 = rp[0], f1 = rp[1], f2 = rp[2], f3 = rp[3];
    float fv[16] = {f0.x, f0.y, f0.z, f0.w, f1.x, f1.y, f1.z, f1.w,
                    f2.x, f2.y, f2.z, f2.w, f3.x, f3.y, f3.z, f3.w};
    float pd[10];
#pragma unroll
    for (int j = 0; j < 10; ++j) {
      float s = 0.f;
#pragma unroll
      for (int i = 0; i < 16; ++i) s += fv[i] * Wl[j][dbase + i];
      pd[j] = s;
    }
    float n2 = 0.f;
#pragma unroll
    for (int i = 0; i < 16; ++i) n2 += fv[i] * fv[i];
#pragma unroll
    for (int off = 16; off >= 1; off >>= 1) {
#pragma unroll
      for (int j = 0; j < 10; ++j) pd[j] += __shfl_xor(pd[j], off, 32);
      n2 += __shfl_xor(n2, off, 32);
    }
    bool valid = (l < length);
    if (valid) {
#pragma unroll
      for (int i = 0; i < 16; ++i) { sum16[i] += fv[i]; sq16[i] += fv[i] * fv[i]; }
    }
    float u[10]; float un2 = 0.f;
#pragma unroll
    for (int j = 0; j < 10; ++j) { u[j] = pd[j] + tbl[j]; un2 += u[j] * u[j]; }
    float rn = 1.0f / fmaxf(sqrtf(un2), 1e-12f);
    if (lane < KALL && valid) {
      float s = 0.f;
#pragma unroll
      for (int j = 0; j < 10; ++j) s += u[j] * rn * tpl[lane][j];
      gacc += s;
    }
    if (lane == 0) wf[F_INVN + rowbase + l] = 1.0f / fmaxf(sqrtf(n2), 1e-12f);
  }
  if (lane < KALL) gspw[wave][lane] = gacc;
#pragma unroll
  for (int i = 0; i < 16; ++i) { red[wave][dbase + i] = sum16[i]; red2[wave][dbase + i] = sq16[i]; }
  __syncthreads();

  float lp = fmaxf((float)length, 1.0f);
  for (int d = tid; d < DIM; d += 256) {
    float s = 0.f, q = 0.f;
#pragma unroll
    for (int w = 0; w < 8; ++w) { s += red[w][d]; q += red2[w][d]; }
    float m = s / lp;
    wf[F_MEAN + (size_t)item * DIM + d] = m;
    wf[F_VAR + (size_t)item * DIM + d] = q / lp - m * m;
  }
  if (tid == 0) {
    float g[KALL];
    for (int c = 0; c < KALL; ++c) {
      float s = 0.f;
      for (int w = 0; w < 8; ++w) s += gspw[w][c];
      g[c] = s;
    }
    for (int k = 0; k < KTOP; ++k) {
      int best = 0; float bv = -3.0e38f;
      for (int c = 0; c < KALL; ++c) if (g[c] > bv) { bv = g[c]; best = c; }
      wi[IW_EXPERT + item * KTOP + k] = best;
      wi[IW_SLOT + item * KTOP + k] = atomicAdd(&wi[IW_COUNTS + best], 1);
      g[best] = -3.0e38f;
    }
  }
}

// ---------------------------------------------------------------------------
// Kernel 2: prefix-sum expert counts -> column offsets
// ---------------------------------------------------------------------------
__global__ void k_prefix(int* __restrict__ wi) {
  if (threadIdx.x == 0) {
    int r = 0;
    for (int e = 0; e < KALL; ++e) { wi[IW_OFFS + e] = r; r += wi[IW_COUNTS + e]; }
  }
}

// ---------------------------------------------------------------------------
// Kernel 3: scatter summaries [mean|var|proxy] into packed expert columns
// ---------------------------------------------------------------------------
__global__ void k_scatter(const float* __restrict__ pp, const float* __restrict__ pg,
                          float* __restrict__ wf, int* __restrict__ wi) {
  __shared__ int cols[KTOP]; __shared__ int exps[KTOP];
  int item = blockIdx.x, tid = threadIdx.x;
  bool isP = item < BP;
  if (tid < KTOP) {
    int e = wi[IW_EXPERT + item * KTOP + tid];
    int c = wi[IW_OFFS + e] + wi[IW_SLOT + item * KTOP + tid];
    wi[IW_COLIDX + item * KTOP + tid] = c;
    cols[tid] = c; exps[tid] = e;
  }
  __syncthreads();
  const float* prox = isP ? pp : pg;
  for (int k = 0; k < KTOP; ++k) {
    float* Sc = wf + F_SMAT + (size_t)cols[k] * 1536;
    const float* pr = prox + (size_t)exps[k] * DIM;
    for (int d = tid; d < DIM; d += 256) {
      Sc[d] = wf[F_MEAN + (size_t)item * DIM + d];
      Sc[DIM + d] = wf[F_VAR + (size_t)item * DIM + d];
      Sc[2 * DIM + d] = pr[d];
    }
  }
}

// ---------------------------------------------------------------------------
// Kernels 4-6: grouped expert GEMM, WMMA f32 16x16x4, double-buffered K=32
// chunks (global loads for chunk i+1 issued before WMMAs of chunk i).
// Out[e][col][m] = leaky( W[e][m][:] . S[e][col][:] + b[e][m] )
// Block: 64(M) x 32(N), 8 waves (4x2), one 16x16 accumulator per wave.
// ---------------------------------------------------------------------------
__global__ void __launch_bounds__(256) k_gemm_expert(
    const float* __restrict__ Wall, const float* __restrict__ Ball,
    const float* __restrict__ Sin, float* __restrict__ Out,
    const int* __restrict__ wi, int M, int K) {
  int e = blockIdx.z;
  int cnt = wi[IW_COUNTS + e];
  if ((int)blockIdx.x * 32 >= cnt) return;
  int col0 = wi[IW_OFFS + e] + blockIdx.x * 32;
  int m0 = blockIdx.y * 64;
  const float* W = Wall + (size_t)e * M * K;

  __shared__ alignas(16) float Wl[64][36];
  __shared__ alignas(16) float Sl[32][36];
  int tid = threadIdx.x;
  int wave = tid >> 5, lane = tid & 31;
  int wm = wave >> 1, wn = wave & 1;
  int l16 = lane & 15, lh = lane >> 4;
  v8f acc = {0.f, 0.f, 0.f, 0.f, 0.f, 0.f, 0.f, 0.f};

  int lrow = tid >> 3;           // 0..31
  int lseg = (tid & 7) * 4;      // 0..28
  const float* Wp0 = W + (size_t)(m0 + lrow) * K;
  const float* Wp1 = W + (size_t)(m0 + lrow + 32) * K;
  const float* Sp  = Sin + (size_t)(col0 + lrow) * K;

  float4 wr0 = *(const float4*)(Wp0 + lseg);
  float4 wr1 = *(const float4*)(Wp1 + lseg);
  float4 sr  = *(const float4*)(Sp + lseg);

  for (int k0 = 0; k0 < K; k0 += 32) {
    // commit staged chunk to LDS
    *(float4*)&Wl[lrow][lseg]      = wr0;
    *(float4*)&Wl[lrow + 32][lseg] = wr1;
    *(float4*)&Sl[lrow][lseg]      = sr;
    __syncthreads();
    // prefetch next chunk into registers (overlaps with WMMA below)
    if (k0 + 32 < K) {
      wr0 = *(const float4*)(Wp0 + k0 + 32 + lseg);
      wr1 = *(const float4*)(Wp1 + k0 + 32 + lseg);
      sr  = *(const float4*)(Sp + k0 + 32 + lseg);
    }
#pragma unroll
    for (int kk = 0; kk < 32; kk += 4) {
      v2f a, b;
      a.x = Wl[wm * 16 + l16][kk + 2 * lh];
      a.y = Wl[wm * 16 + l16][kk + 2 * lh + 1];
      b.x = Sl[wn * 16 + l16][kk + 2 * lh];
      b.y = Sl[wn * 16 + l16][kk + 2 * lh + 1];
      acc = __builtin_amdgcn_wmma_f32_16x16x4_f32(false, a, false, b, (short)0, acc, false, false);
    }
    __syncthreads();
  }
  int n = col0 + wn * 16 + l16;
  int mb = m0 + wm * 16 + 8 * lh;
  float* oc = Out + (size_t)n * M + mb;
  const float* bb = Ball + (size_t)e * M + mb;
#pragma unroll
  for (int r = 0; r < 8; ++r) {
    float v = acc[r] + bb[r];
    oc[r] = (v > 0.f) ? v : 0.01f * v;
  }
}

// ---------------------------------------------------------------------------
// Kernel 7: gather probe ctx into row-major pack + beta = probe_b . ctx
// ---------------------------------------------------------------------------
__global__ void k_pack(const float* __restrict__ probe_b, float* __restrict__ wf,
                       const int* __restrict__ wi) {
  __shared__ float ctxs[KTOP][DIM];
  int item = blockIdx.x, tid = threadIdx.x;
  for (int k = 0; k < KTOP; ++k) {
    int col = wi[IW_COLIDX + item * KTOP + k];
    for (int d = tid; d < DIM; d += 256) {
      float v = wf[F_CTX + (size_t)col * DIM + d];
      ctxs[k][d] = v;
      wf[F_CTXP + ((size_t)item * KTOP + k) * DIM + d] = v;
    }
  }
  __syncthreads();
  int wave = tid >> 5, lane = tid & 31;
  if (wave < KTOP) {
    float s = 0.f;
#pragma unroll
    for (int i = 0; i < 16; ++i) s += ctxs[wave][lane * 16 + i] * probe_b[lane * 16 + i];
    s = wred_sum(s);
    if (lane == 0) wf[F_BETA + item * KTOP + wave] = s;
  }
}

// ---------------------------------------------------------------------------
// Kernel 8: Ctilde[p][i] = sum_o CtxP[p][o] * probe_W[o][i]  (512x512x512 GEMM)
// Double-buffered like the expert GEMM; B is row-major K x N.
// ---------------------------------------------------------------------------
__global__ void __launch_bounds__(256) k_gemm_ct(const float* __restrict__ A,
                                                 const float* __restrict__ Bw,
                                                 float* __restrict__ C) {
  const int N = 512, K = 512;
  int n0 = blockIdx.x * 32, m0 = blockIdx.y * 64;
  __shared__ alignas(16) float Al[64][36];
  __shared__ alignas(16) float Bl[32][36];
  int tid = threadIdx.x, wave = tid >> 5, lane = tid & 31;
  int wm = wave >> 1, wn = wave & 1, l16 = lane & 15, lh = lane >> 4;
  v8f acc = {0.f, 0.f, 0.f, 0.f, 0.f, 0.f, 0.f, 0.f};

  int lrow = tid >> 3;           // 0..31
  int lseg = (tid & 7) * 4;      // 0..28
  const float* Ap0 = A + (size_t)(m0 + lrow) * K;
  const float* Ap1 = A + (size_t)(m0 + lrow + 32) * K;

  float4 ar0 = *(const float4*)(Ap0 + lseg);
  float4 ar1 = *(const float4*)(Ap1 + lseg);
  float4 br  = *(const float4*)(Bw + (size_t)lrow * N + n0 + lseg);

  for (int k0 = 0; k0 < K; k0 += 32) {
    *(float4*)&Al[lrow][lseg]      = ar0;
    *(float4*)&Al[lrow + 32][lseg] = ar1;
    *(float4*)&Bl[lrow][lseg]      = br;
    __syncthreads();
    if (k0 + 32 < K) {
      ar0 = *(const float4*)(Ap0 + k0 + 32 + lseg);
      ar1 = *(const float4*)(Ap1 + k0 + 32 + lseg);
      br  = *(const float4*)(Bw + (size_t)(k0 + 32 + lrow) * N + n0 + lseg);
    }
#pragma unroll
    for (int kk = 0; kk < 32; kk += 4) {
      v2f a, b;
      a.x = Al[wm * 16 + l16][kk + 2 * lh];
      a.y = Al[wm * 16 + l16][kk + 2 * lh + 1];
      b.x = Bl[kk + 2 * lh][wn * 16 + l16];
      b.y = Bl[kk + 2 * lh + 1][wn * 16 + l16];
      acc = __builtin_amdgcn_wmma_f32_16x16x4_f32(false, a, false, b, (short)0, acc, false, false);
    }
    __syncthreads();
  }
  int n = n0 + wn * 16 + l16;
  int mb = m0 + wm * 16 + 8 * lh;
#pragma unroll
  for (int r = 0; r < 8; ++r) C[(size_t)(mb + r) * N + n] = acc[r];
}

// ---------------------------------------------------------------------------
// Kernel 9: per-item scores -> softmax over set -> weighted sum of normed rows
// ---------------------------------------------------------------------------
__global__ void __launch_bounds__(256) k_final(
    const float* __restrict__ probes, const float* __restrict__ gallery,
    const int* __restrict__ plen, const int* __restrict__ glen,
    float* __restrict__ out, const float* __restrict__ wf, const int* __restrict__ wi) {
  __shared__ float ctxs[KTOP][DIM];
  __shared__ float bet[KTOP];
  __shared__ float sarr[LG * KTOP];
  __shared__ float mx[KTOP], rden[KTOP];
  __shared__ float obuf[4][KTOP * DIM];

  int item = blockIdx.x, tid = threadIdx.x;
  bool isP = item < BP;
  const float* feat = isP ? probes + (size_t)item * LP * DIM
                          : gallery + (size_t)(item - BP) * LG * DIM;
  int L = isP ? LP : LG;
  int length = isP ? plen[item] : glen[item - BP];
  size_t rowbase = isP ? (size_t)item * LP : (size_t)BP * LP + (size_t)(item - BP) * LG;

  if (isP) {
    for (int i = tid; i < KTOP * DIM; i += 256)
      ctxs[i / DIM][i % DIM] = wf[F_CTILDE + (size_t)item * KTOP * DIM + i];
    if (tid < KTOP) bet[tid] = wf[F_BETA + item * KTOP + tid];
  } else {
    for (int k = 0; k < KTOP; ++k) {
      int col = wi[IW_COLIDX + item * KTOP + k];
      for (int d = tid; d < DIM; d += 256) ctxs[k][d] = wf[F_CTX + (size_t)col * DIM + d];
    }
    if (tid < KTOP) bet[tid] = 0.f;
  }
  __syncthreads();

  int wave = tid >> 5, lane = tid & 31, dbase = lane * 16;
  for (int l = wave; l < L; l += 8) {
    const float4* rp = (const float4*)(feat + (size_t)l * DIM + dbase);
    float4 f0 = rp[0], f1 = rp[1], f2 = rp[2], f3 = rp[3];
    float fv[16] = {f0.x, f0.y, f0.z, f0.w, f1.x, f1.y, f1.z, f1.w,
                    f2.x, f2.y, f2.z, f2.w, f3.x, f3.y, f3.z, f3.w};
    float p[KTOP];
#pragma unroll
    for (int k = 0; k < KTOP; ++k) {
      float s = 0.f;
#pragma unroll
      for (int i = 0; i < 16; ++i) s += fv[i] * ctxs[k][dbase + i];
      p[k] = s;
    }
#pragma unroll
    for (int off = 16; off >= 1; off >>= 1) {
#pragma unroll
      for (int k = 0; k < KTOP; ++k) p[k] += __shfl_xor(p[k], off, 32);
    }
    if (lane < KTOP) {
      float sc = (l < length) ? p[lane] + bet[lane] : NEG_INF_F;
      sarr[l * KTOP + lane] = sc;
    }
  }
  __syncthreads();

  if (wave < KTOP) {
    int k = wave;
    float m = -3.0e38f;
    for (int l = lane; l < L; l += 32) m = fmaxf(m, sarr[l * KTOP + k]);
    m = wred_max(m);
    float s = 0.f;
    for (int l = lane; l < L; l += 32) s += expf(sarr[l * KTOP + k] - m);
    s = wred_sum(s);
    if (lane == 0) { mx[k] = m; rden[k] = 1.0f / s; }
  }
  __syncthreads();

  float acc[KTOP][16];
#pragma unroll
  for (int k = 0; k < KTOP; ++k)
#pragma unroll
    for (int i = 0; i < 16; ++i) acc[k][i] = 0.f;

  for (int l = wave; l < L; l += 8) {
    const float4* rp = (const float4*)(feat + (size_t)l * DIM + dbase);
    float4 f0 = rp[0], f1 = rp[1], f2 = rp[2], f3 = rp[3];
    float fv[16] = {f0.x, f0.y, f0.z, f0.w, f1.x, f1.y, f1.z, f1.w,
                    f2.x, f2.y, f2.z, f2.w, f3.x, f3.y, f3.z, f3.w};
    float inr = wf[F_INVN + rowbase + l];
    float w[KTOP];
#pragma unroll
    for (int k = 0; k < KTOP; ++k) w[k] = expf(sarr[l * KTOP + k] - mx[k]) * rden[k] * inr;
#pragma unroll
    for (int k = 0; k < KTOP; ++k)
#pragma unroll
      for (int i = 0; i < 16; ++i) acc[k][i] += fv[i] * w[k];
  }
  if (wave < 4) {
#pragma unroll
    for (int k = 0; k < KTOP; ++k)
#pragma unroll
      for (int i = 0; i < 16; ++i) obuf[wave][k * DIM + dbase + i] = acc[k][i];
  }
  __syncthreads();
  if (wave >= 4) {
#pragma unroll
    for (int k = 0; k < KTOP; ++k)
#pragma unroll
      for (int i = 0; i < 16; ++i) obuf[wave - 4][k * DIM + dbase + i] += acc[k][i];
  }
  __syncthreads();

  float* op = out + (isP ? (size_t)item * KTOP * DIM
                         : (size_t)OUT_FG + (size_t)(item - BP) * KTOP * DIM);
  for (int t = tid; t < KTOP * DIM; t += 256)
    op[t] = obuf[0][t] + obuf[1][t] + obuf[2][t] + obuf[3][t];
}

// ---------------------------------------------------------------------------
extern "C" void kernel_launch(void* const* d_in, const int* in_sizes, int n_in,
                              void* d_out, int out_size, void* d_ws, size_t ws_size,
                              hipStream_t stream) {
  const float* probes  = (const float*)d_in[0];
  const float* gallery = (const float*)d_in[1];
  const int*   plen    = (const int*)d_in[2];
  const int*   glen    = (const int*)d_in[3];
  const float* pp      = (const float*)d_in[4];
  const float* pg      = (const float*)d_in[5];
  const float* tW      = (const float*)d_in[6];
  const float* tb      = (const float*)d_in[7];
  const float* probe_W = (const float*)d_in[8];
  const float* probe_b = (const float*)d_in[9];
  const float* W1      = (const float*)d_in[10];
  const float* b1      = (const float*)d_in[11];
  const float* W2      = (const float*)d_in[12];
  const float* b2      = (const float*)d_in[13];
  const float* W3      = (const float*)d_in[14];
  const float* b3      = (const float*)d_in[15];
  float* out = (float*)d_out;
  float* wf  = (float*)d_ws;
  int*   wi  = (int*)d_ws;

  k_init<<<1, 256, 0, stream>>>(pp, pg, tW, tb, out, wf, wi);
  k_stats<<<NITEMS, 256, 0, stream>>>(probes, gallery, plen, glen, tW, tb, wf, wi);
  k_prefix<<<1, 32, 0, stream>>>(wi);
  k_scatter<<<NITEMS, 256, 0, stream>>>(pp, pg, wf, wi);
  k_gemm_expert<<<dim3(6, 16, KALL), 256, 0, stream>>>(W1, b1, wf + F_SMAT, wf + F_H1, wi, 1024, 1536);
  k_gemm_expert<<<dim3(6, 16, KALL), 256, 0, stream>>>(W2, b2, wf + F_H1, wf + F_H2, wi, 1024, 1024);
  k_gemm_expert<<<dim3(6, 8, KALL), 256, 0, stream>>>(W3, b3, wf + F_H2, wf + F_CTX, wi, 512, 1024);
  k_pack<<<BP, 256, 0, stream>>>(probe_b, wf, wi);
  k_gemm_ct<<<dim3(16, 8), 256, 0, stream>>>(wf + F_CTXP, probe_W, wf + F_CTILDE);
  k_final<<<NITEMS, 256, 0, stream>>>(probes, gallery, plen, glen, out, wf, wi);
}